// CPDBlock_66236985639229
// MI455X (gfx1250) — compile-verified
//
#include <hip/hip_runtime.h>

// ---------------------------------------------------------------------------
// CPD block on MI455X (gfx1250, wave32, WMMA).
// 66 GFLOP fp32 vs ~0.5 GB HBM @ 23.3 TB/s => compute bound; use
// V_WMMA_F32_16X16X4_F32 (exact fp32) for all three GEMM-shaped stages.
// Block tile 64x256 (8 waves, 2Mx4N), wave tile 32x64 (8 v8f accumulators).
// A panel [64][32] and B panel [256][32] both stored k-major, stride 34 =>
// every WMMA fragment is a single aligned ds_load_b64, no shuffle movs.
// Register double-buffering: global loads for chunk c+1 issue before the
// WMMA loop over chunk c.
// ---------------------------------------------------------------------------

typedef float v2f __attribute__((ext_vector_type(2)));
typedef float v8f __attribute__((ext_vector_type(8)));

#define B_    16
#define CIN_  256
#define COUT_ 256
#define RANK_ 64
#define KSZ   3
#define H_    112
#define W_    112
#define HW_   (H_ * W_)          // 12544 = 49 * 256 (exact N tiling)

// Workspace layout (float elements)
#define OFF_WSUM 0
#define OFF_WB   (OFF_WSUM + COUT_ * CIN_)          // 65536
#define OFF_WT   (OFF_WB + KSZ * RANK_ * COUT_)     // +49152
#define OFF_Y1   (OFF_WT + KSZ * COUT_ * RANK_)     // +49152
#define OFF_Y2   (OFF_Y1 + B_ * COUT_ * HW_)        // +51380224
// total = 64,389,120 floats (~258 MB) in d_ws

static __device__ inline v8f wmma4(v2f a, v2f b, v8f c) {
  // (neg_a, A, neg_b, B, c_mod, C, reuse_a, reuse_b)
  return __builtin_amdgcn_wmma_f32_16x16x4_f32(false, a, false, b,
                                               (short)0, c, false, false);
}

// ---------------------------------------------------------------------------
// Prep: rank-sum w_head -> W_sum[o][i]; repack w_body -> [kh][r][o],
// w_tail -> [kw][c][r] so GEMM A-panels are contiguous.
// ---------------------------------------------------------------------------
__global__ __launch_bounds__(256) void cpd_prep(const float* __restrict__ wh,
                                                const float* __restrict__ wb,
                                                const float* __restrict__ wt,
                                                float* __restrict__ ws) {
  int gid = blockIdx.x * 256 + threadIdx.x;
  if (gid < COUT_ * CIN_) {
    int o = gid >> 8, i = gid & 255;
    float s = 0.f;
#pragma unroll 4
    for (int r = 0; r < RANK_; ++r) s += wh[(r * COUT_ + o) * CIN_ + i];
    ws[OFF_WSUM + gid] = s;
  } else if (gid < COUT_ * CIN_ + KSZ * RANK_ * COUT_) {
    int idx = gid - COUT_ * CIN_;
    int kh = idx / (RANK_ * COUT_);
    int rem = idx - kh * RANK_ * COUT_;
    int r = rem >> 8, o = rem & 255;
    ws[OFF_WB + idx] = wb[(r * COUT_ + o) * KSZ + kh];       // [RANK,COUT,3,1]
  } else if (gid < COUT_ * CIN_ + 2 * KSZ * RANK_ * COUT_) {
    int idx = gid - COUT_ * CIN_ - KSZ * RANK_ * COUT_;
    int kw = idx / (COUT_ * RANK_);
    int rem = idx - kw * COUT_ * RANK_;
    int c = rem >> 6, r = rem & 63;
    ws[OFF_WT + idx] = wt[(c * RANK_ + r) * KSZ + kw];       // [COUT,RANK,1,3]
  }
}

// ---------------------------------------------------------------------------
// Unified shifted-GEMM stage.
//   MODE 0: no mask, float4 B loads           (stage 1, 1x1 conv)
//   MODE 1: flat-range mask, shift=(tap-1)*W  (stage 2, (3,1) conv, pad H)
//   MODE 2: w-coord mask,   shift=(tap-1)*1   (stage 3, (1,3) conv, pad W)
// Out[b, m, p] = sum_{tap, k} A[tap][m][k] * Bsrc[b, k, p + shift(tap)]
// ---------------------------------------------------------------------------
template <int MODE, int BIAS, int T, int KA, int BC, int SHIFTU, int MTOT>
__global__ __launch_bounds__(256) void cpd_gemm(const float* __restrict__ Bsrc,
                                                const float* __restrict__ Apack,
                                                const float* __restrict__ bias,
                                                float* __restrict__ Out) {
  constexpr int NCpT = KA / 32;
  constexpr int NCHUNKS = T * NCpT;

  __shared__ float As[64 * 34];    // [m 0..63][k 0..31], k-major, stride 34
  __shared__ float Bs[256 * 34];   // [col 0..255][k 0..31], k-major, stride 34

  const int tid = threadIdx.x;
  const int lane = tid & 31;
  const int wave = tid >> 5;
  const int lh = lane >> 4, l15 = lane & 15;
  const int wm = wave >> 2;          // 0..1  (M)
  const int wn = wave & 3;           // 0..3  (N)
  const int pbase = blockIdx.x * 256;
  const int m0 = blockIdx.y * 64;
  const int b = blockIdx.z;

  v8f acc[2][4] = {};
  float areg[8];
  float4 breg[8];   // MODE 0
  float sreg[32];   // MODE 1/2

  auto loadRegs = [&](int c) {
    const int tap = c / NCpT;
    const int kb = (c % NCpT) * 32;
    const float* Ag = Apack + ((size_t)tap * MTOT + m0) * KA + kb;
#pragma unroll
    for (int q = 0; q < 8; ++q) {
      int idx = tid + q * 256;
      int r = idx >> 5, kk = idx & 31;
      areg[q] = Ag[r * KA + kk];
    }
    if (MODE == 0) {
#pragma unroll
      for (int q = 0; q < 8; ++q) {
        int idx = tid + q * 256;
        int r = idx >> 6, c4 = idx & 63;
        const float4* src =
            (const float4*)(Bsrc + ((size_t)b * BC + kb + r) * HW_ + pbase);
        breg[q] = src[c4];
      }
    } else {
      const int off = (tap - 1) * SHIFTU;
#pragma unroll
      for (int q = 0; q < 32; ++q) {
        int idx = tid + q * 256;
        int r = idx >> 8, j = idx & 255;
        int p = pbase + j;
        bool ok;
        if (MODE == 1) {          // whole-row shift: flat range check
          int pos = p + off;
          ok = (pos >= 0) && (pos < HW_);
        } else {                  // +-1 shift: check w coordinate
          int w = p - (p / W_) * W_;
          int wp = w + off;
          ok = (wp >= 0) && (wp < W_);
        }
        float v = 0.f;
        if (ok) v = Bsrc[((size_t)b * BC + kb + r) * HW_ + p + off];
        sreg[q] = v;
      }
    }
  };

  auto storeRegs = [&]() {
#pragma unroll
    for (int q = 0; q < 8; ++q) {
      int idx = tid + q * 256;
      int r = idx >> 5, kk = idx & 31;
      As[r * 34 + kk] = areg[q];
    }
    if (MODE == 0) {
#pragma unroll
      for (int q = 0; q < 8; ++q) {
        int idx = tid + q * 256;
        int r = idx >> 6, c4 = idx & 63;
        Bs[(c4 * 4 + 0) * 34 + r] = breg[q].x;
        Bs[(c4 * 4 + 1) * 34 + r] = breg[q].y;
        Bs[(c4 * 4 + 2) * 34 + r] = breg[q].z;
        Bs[(c4 * 4 + 3) * 34 + r] = breg[q].w;
      }
    } else {
#pragma unroll
      for (int q = 0; q < 32; ++q) {
        int idx = tid + q * 256;
        int r = idx >> 8, j = idx & 255;
        Bs[j * 34 + r] = sreg[q];   // stride-34 scatter: conflict-free
      }
    }
  };

  loadRegs(0);
  const int ar0 = (wm * 32 + l15) * 34 + 2 * lh;
  const int ar1 = (wm * 32 + 16 + l15) * 34 + 2 * lh;
  const int br = (wn * 64 + l15) * 34 + 2 * lh;

  for (int c = 0; c < NCHUNKS; ++c) {
    storeRegs();
    __syncthreads();
    if (c + 1 < NCHUNKS) loadRegs(c + 1);   // global latency hides under WMMA
#pragma unroll
    for (int k = 0; k < 32; k += 4) {
      v2f a0 = *(const v2f*)&As[ar0 + k];   // ds_load_b64
      v2f a1 = *(const v2f*)&As[ar1 + k];
#pragma unroll
      for (int nt = 0; nt < 4; ++nt) {
        v2f bf = *(const v2f*)&Bs[br + nt * 16 * 34 + k];  // ds_load_b64
        acc[0][nt] = wmma4(a0, bf, acc[0][nt]);
        acc[1][nt] = wmma4(a1, bf, acc[1][nt]);
      }
    }
    __syncthreads();
  }

#pragma unroll
  for (int mt = 0; mt < 2; ++mt) {
#pragma unroll
    for (int nt = 0; nt < 4; ++nt) {
      int col = pbase + wn * 64 + nt * 16 + l15;
#pragma unroll
      for (int v = 0; v < 8; ++v) {
        int row = m0 + wm * 32 + mt * 16 + v + 8 * lh;
        float o = acc[mt][nt][v];
        if (BIAS) o += bias[row];
        Out[((size_t)b * MTOT + row) * HW_ + col] = o;
      }
    }
  }
}

// ---------------------------------------------------------------------------
extern "C" void kernel_launch(void* const* d_in, const int* in_sizes, int n_in,
                              void* d_out, int out_size, void* d_ws,
                              size_t ws_size, hipStream_t stream) {
  const float* x  = (const float*)d_in[0];  // [16,256,112,112]
  const float* wh = (const float*)d_in[1];  // [64,256,256]
  const float* wb = (const float*)d_in[2];  // [64,256,3,1]
  const float* wt = (const float*)d_in[3];  // [256,64,1,3]
  const float* bt = (const float*)d_in[4];  // [256]
  float* ws  = (float*)d_ws;
  float* out = (float*)d_out;

  (void)in_sizes; (void)n_in; (void)out_size; (void)ws_size;

  cpd_prep<<<640, 256, 0, stream>>>(wh, wb, wt, ws);

  // Stage 1: Y1 = Wsum(256x256) * X  | M=256 K=256 N=12544 per batch
  cpd_gemm<0, 0, 1, 256, 256, 0, 256>
      <<<dim3(HW_ / 256, 4, B_), 256, 0, stream>>>(
          x, ws + OFF_WSUM, nullptr, ws + OFF_Y1);

  // Stage 2: Y2 = (3,1)-conv 256->64, pad H | 3 taps, K=256, M=64
  cpd_gemm<1, 0, 3, 256, 256, W_, 64>
      <<<dim3(HW_ / 256, 1, B_), 256, 0, stream>>>(
          ws + OFF_Y1, ws + OFF_WB, nullptr, ws + OFF_Y2);

  // Stage 3: Out = (1,3)-conv 64->256 + bias, pad W | 3 taps, K=64, M=256
  cpd_gemm<2, 1, 3, 64, 64, 1, 256>
      <<<dim3(HW_ / 256, 4, B_), 256, 0, stream>>>(
          ws + OFF_Y2, ws + OFF_WT, bt, out);
}